// LaplacianLoss_19559281066265
// MI455X (gfx1250) — compile-verified
//
#include <hip/hip_runtime.h>
#include <hip/hip_bf16.h>

// Laplacian loss, N=8192, D=3.
// loss = sum || d - (A@d)/deg ||^2 , d = pred2-pred1, deg = column sums of A.
// Bandwidth-bound: 2 passes over the 268MB fp32 A matrix (~23us @ 23.3 TB/s).
// Matvec pass uses V_WMMA_F32_16X16X4_F32 (wave32, gfx1250).

typedef __attribute__((ext_vector_type(2))) float v2f;
typedef __attribute__((ext_vector_type(8))) float v8f;

#define NROWS   8192
#define DDIM    3
#define NELEM   (NROWS * DDIM)      // 24576
#define KCHUNK  512                 // columns per K-chunk
#define NKCH    (NROWS / KCHUNK)    // 16 K-chunks
#define RCHUNK  512                 // rows per deg partial chunk
#define NRCH    (NROWS / RCHUNK)    // 16 deg partials

// ---------------------------------------------------------------- kernel 1
__global__ void __launch_bounds__(256)
lap_prep(const float* __restrict__ p1, const float* __restrict__ p2,
         float* __restrict__ dd) {
    int i = blockIdx.x * 256 + threadIdx.x;
    if (i < NELEM) dd[i] = p2[i] - p1[i];
}

// ---------------------------------------------------------------- kernel 2
// Partial column sums of A, deterministic (no atomics).
// grid (32, 16): x = 256-column block, y = 512-row chunk. Coalesced across j.
__global__ void __launch_bounds__(256)
lap_deg_partial(const float* __restrict__ A, float* __restrict__ degP) {
    int j  = blockIdx.x * 256 + threadIdx.x;
    int rc = blockIdx.y;
    const float* p = A + (size_t)rc * RCHUNK * NROWS + j;
    float s = 0.0f;
    #pragma unroll 8
    for (int r = 0; r < RCHUNK; ++r) s += p[(size_t)r * NROWS];
    degP[rc * NROWS + j] = s;
}

// ---------------------------------------------------------------- kernel 3
// y = A @ d via V_WMMA_F32_16X16X4_F32.
// grid (64, 16), 256 threads = 8 waves. Wave w of block (bx, kc) computes the
// partial matvec for rows [ (bx*8+w)*16, +16 ) over columns [kc*512, +512),
// writing a 16x3 partial into yP[kc].
//
// A-matrix fragment layout (ISA 7.12.2, 32-bit A 16x4): lane L holds row
// M = L&15; lanes 0-15 carry K=0,1 and lanes 16-31 carry K=2,3 in VGPR0/1
// -> one global_load_b64 of A[row, k + (L>=16 ? 2:0)] per tile per lane.
// B-matrix (4x16, d-values): lane L = column N = L&15; VGPR0/1 hold rows
// K = (L>>4)*2 and +1 -> one ds_load_b64 from dT[n][k] (transposed d in LDS).
// dT padded to 4 rows (row 3 = 0) so (col&3) indexing is always in-bounds;
// B columns 3..15 only feed C columns we never read.
__global__ void __launch_bounds__(256)
lap_wmma_matvec(const float* __restrict__ A, const float* __restrict__ dd,
                float* __restrict__ yP) {
    __shared__ float dT[4 * KCHUNK];

    const int tid = threadIdx.x;
    const int kc  = blockIdx.y;
    const int k0  = kc * KCHUNK;

    // Stage d chunk transposed: dT[n][k] = d[k0+k][n]; row 3 zeroed.
    for (int idx = tid; idx < 3 * KCHUNK; idx += 256) {
        int k = idx / 3, n = idx - 3 * k;
        dT[n * KCHUNK + k] = dd[(k0 + k) * DDIM + n];
    }
    for (int idx = tid; idx < KCHUNK; idx += 256) dT[3 * KCHUNK + idx] = 0.0f;
    __syncthreads();

    const int wave = tid >> 5;
    const int lane = tid & 31;
    const int col  = lane & 15;        // M for A-frag, N for B/C-frag
    const int hi   = lane >> 4;        // 0 | 1
    const int kr   = hi << 1;          // K offset 0 | 2
    const int r0   = (blockIdx.x * 8 + wave) * 16;

    const float* aRow = A + (size_t)(r0 + col) * NROWS + k0 + kr;
    const float* bRow = dT + (col & 3) * KCHUNK + kr;

    v8f c = {};
    #pragma unroll 8
    for (int j = 0; j < KCHUNK; j += 4) {
        v2f a = *(const v2f*)(aRow + j);   // A[row, k..k+1] (global_load_b64)
        v2f b = *(const v2f*)(bRow + j);   // dT[n][k..k+1]  (ds_load_b64)
        // D = A(16x4,f32) x B(4x16,f32) + C
        c = __builtin_amdgcn_wmma_f32_16x16x4_f32(
                false, a, false, b, (short)0, c, false, false);
    }

    // C/D layout: VGPR v -> (M = v + hi*8, N = col). Keep N = 0..2 only.
    if (col < 3) {
        float* out = yP + (size_t)kc * NELEM + (r0 + hi * 8) * DDIM + col;
        #pragma unroll
        for (int v = 0; v < 8; ++v) out[v * DDIM] = c[v];
    }
}

// ---------------------------------------------------------------- kernel 4
// Deterministic finalize: deg/y partial sums in fixed order, single-block
// tree reduction of sum((d - y/deg)^2).
__global__ void __launch_bounds__(1024)
lap_finalize(const float* __restrict__ dd, const float* __restrict__ degP,
             const float* __restrict__ yP, float* __restrict__ out) {
    __shared__ float red[1024];
    int tid = threadIdx.x;
    float acc = 0.0f;
    for (int i = tid; i < NELEM; i += 1024) {
        int row = i / DDIM;
        float deg = 0.0f;
        #pragma unroll
        for (int p = 0; p < NRCH; ++p) deg += degP[p * NROWS + row];
        float y = 0.0f;
        #pragma unroll
        for (int p = 0; p < NKCH; ++p) y += yP[(size_t)p * NELEM + i];
        float r = dd[i] - y / deg;
        acc += r * r;
    }
    red[tid] = acc;
    __syncthreads();
    for (int s = 512; s > 0; s >>= 1) {
        if (tid < s) red[tid] += red[tid + s];
        __syncthreads();
    }
    if (tid == 0) out[0] = red[0];
}

// ---------------------------------------------------------------- launch
extern "C" void kernel_launch(void* const* d_in, const int* in_sizes, int n_in,
                              void* d_out, int out_size, void* d_ws, size_t ws_size,
                              hipStream_t stream) {
    const float* pred1 = (const float*)d_in[0];
    const float* pred2 = (const float*)d_in[1];
    const float* A     = (const float*)d_in[2];

    float* ws   = (float*)d_ws;
    float* dd   = ws;                                   // 24576 floats
    float* degP = ws + NELEM;                           // 16*8192 floats
    float* yP   = ws + NELEM + NRCH * NROWS;            // 16*24576 floats

    lap_prep<<<(NELEM + 255) / 256, 256, 0, stream>>>(pred1, pred2, dd);
    lap_deg_partial<<<dim3(NROWS / 256, NRCH), 256, 0, stream>>>(A, degP);
    lap_wmma_matvec<<<dim3(NROWS / (8 * 16), NKCH), 256, 0, stream>>>(A, dd, yP);
    lap_finalize<<<1, 1024, 0, stream>>>(dd, degP, yP, (float*)d_out);
}